// ChamferDistanceLoss_68143951118336
// MI455X (gfx1250) — compile-verified
//
#include <hip/hip_runtime.h>
#include <hip/hip_bf16.h>
#include <math.h>

// Chamfer distance, MI455X (gfx1250), wave32.
//
// d(x,y)^2 = |x|^2 + |y|^2 - 2 x.y  via V_WMMA_F32_16X16X4_F32:
//   A (16x4 f32) rows = (-2x0, -2x1, -2x2, 1)       (one X point per row)
//   B (4x16 f32) cols = ( y0,   y1,   y2, |y|^2)    (one Y point per col)
//   D = |y|^2 - 2 x.y   (C = inline 0, hoisted)
// |x|^2 is row-constant under min-over-columns -> added after the min
// reduction, then clamp + sqrt.
//
// ISA operand layouts (CDNA5 ISA, 32-bit 16x4 A / 16x16 C):
//   A: VGPR0 = K0 (lanes 0-15) | K2 (lanes 16-31); VGPR1 = K1 | K3
//   B: mirrored (N on lanes):  VGPR0 = K0 | K2;    VGPR1 = K1 | K3
//   D: VGPR r = row M=r (lanes 0-15, N=lane) | M=r+8 (lanes 16-31, N=lane-16)
//
// LDS staging is SPLIT by lane half so the inner loop needs no cndmask:
//   loPts[p] = (y0, y1)        read by lanes 0-15  (K0,K1)
//   hiPts[p] = (y2, |y|^2)     read by lanes 16-31 (K2,K3)
// hiPts is offset by 32 dwords so the two halves hit disjoint LDS bank sets
// (otherwise both halves land on the same 32 banks -> 2-way conflict).

typedef __attribute__((ext_vector_type(2))) float v2f;
typedef __attribute__((ext_vector_type(8))) float v8f;

#define NPTS   8192
#define CHUNK  4096            // Y points per LDS stage
#define ROWS_PER_BLOCK 128     // 8 waves * 16 rows
#define LDS_FLOATS (4 * CHUNK + 32)   // lo[2*CHUNK] + 32-float pad + hi[2*CHUNK]

__global__ void chamfer_zero_kernel(float* out) { out[0] = 0.0f; }

__global__ __launch_bounds__(256) void chamfer_dir_kernel(
    const float* __restrict__ A, const float* __restrict__ B,
    float* __restrict__ out)
{
    __shared__ float sY[LDS_FLOATS];
    float2* loPts = (float2*)sY;                       // [CHUNK] = (y0, y1)
    float2* hiPts = (float2*)(sY + 2 * CHUNK + 32);    // [CHUNK] = (y2, |y|^2)

    const int tid   = threadIdx.x;
    const int lane  = tid & 31;
    const int wave  = tid >> 5;          // 0..7
    const int lhalf = lane & 15;
    const bool hi   = lane >= 16;

    const int batch = blockIdx.y;        // 0..3
    const int dir   = blockIdx.z;        // 0: A->B mins, 1: B->A mins

    const float* X = (dir ? B : A) + (size_t)batch * (NPTS * 3);
    const float* Y = (dir ? A : B) + (size_t)batch * (NPTS * 3);

    // ---- This wave owns X rows [rowBase, rowBase+16). Lane L holds point
    // rowBase + (L & 15); both lane halves load the same point (different K).
    const int rowBase = (blockIdx.x * 8 + wave) * 16;
    const int myRow   = rowBase + lhalf;
    const float x0 = X[myRow * 3 + 0];
    const float x1 = X[myRow * 3 + 1];
    const float x2 = X[myRow * 3 + 2];
    const float xsq = x0 * x0 + x1 * x1 + x2 * x2;

    // A operand: lanes 0-15 -> (K0,K1)=(-2x0,-2x1); lanes 16-31 -> (K2,K3)=(-2x2, 1)
    v2f a;
    a.x = hi ? (-2.0f * x2) : (-2.0f * x0);
    a.y = hi ? 1.0f         : (-2.0f * x1);

    // |x|^2 broadcast matching the D layout: accumulator r needs row r (+8 for hi half).
    float xsqr[8];
#pragma unroll
    for (int r = 0; r < 8; ++r)
        xsqr[r] = __shfl(xsq, r + (hi ? 8 : 0), 32);

    // Loop-invariant per-lane LDS base: this lane's half, its column slot.
    const float2* lanePtr = (hi ? hiPts : loPts) + lhalf;

    v8f mn = {3e38f, 3e38f, 3e38f, 3e38f, 3e38f, 3e38f, 3e38f, 3e38f};

    for (int chunk = 0; chunk < 2; ++chunk) {
        __syncthreads();  // previous stage's consumers done before overwrite
        const float* Yc = Y + (size_t)(chunk * CHUNK) * 3;
        for (int p = tid; p < CHUNK; p += 256) {
            float y0 = Yc[p * 3 + 0];
            float y1 = Yc[p * 3 + 1];
            float y2 = Yc[p * 3 + 2];
            loPts[p] = make_float2(y0, y1);
            hiPts[p] = make_float2(y2, y0 * y0 + y1 * y1 + y2 * y2);
        }
        __syncthreads();

        // 16 Y columns per WMMA tile: ds_load_b64 -> wmma -> min accumulate.
#pragma unroll 4
        for (int t = 0; t < CHUNK / 16; ++t) {
            float2 yv = lanePtr[t * 16];
            v2f bv;
            bv.x = yv.x;
            bv.y = yv.y;
            v8f d = __builtin_amdgcn_wmma_f32_16x16x4_f32(
                false, a, false, bv, (short)0,
                (v8f){0.f, 0.f, 0.f, 0.f, 0.f, 0.f, 0.f, 0.f}, false, false);
#pragma unroll
            for (int r = 0; r < 8; ++r)
                mn[r] = fminf(mn[r], d[r]);
        }
    }

    // ---- Reduce mins across the 16 lanes of each half (columns), finish rows.
    float sum = 0.0f;
#pragma unroll
    for (int r = 0; r < 8; ++r) {
        float v = mn[r];
        v = fminf(v, __shfl_xor(v, 1, 32));
        v = fminf(v, __shfl_xor(v, 2, 32));
        v = fminf(v, __shfl_xor(v, 4, 32));
        v = fminf(v, __shfl_xor(v, 8, 32));
        sum += sqrtf(fmaxf(v + xsqr[r], 0.0f));  // row min distance
    }
    sum += __shfl_xor(sum, 16, 32);  // combine the two 8-row halves

    // result = sum_over_everything(min dists) / (4 * 8192 * 12.8); 10/2^22 is exact.
    if (lane == 0)
        atomicAdd(out, sum * (10.0f / 4194304.0f));
}

extern "C" void kernel_launch(void* const* d_in, const int* in_sizes, int n_in,
                              void* d_out, int out_size, void* d_ws, size_t ws_size,
                              hipStream_t stream) {
    const float* A = (const float*)d_in[0];
    const float* B = (const float*)d_in[1];
    float* out = (float*)d_out;

    chamfer_zero_kernel<<<1, 1, 0, stream>>>(out);

    // grid: x = row tiles (8192 / 128), y = batch, z = direction
    dim3 grid(NPTS / ROWS_PER_BLOCK, 4, 2);
    chamfer_dir_kernel<<<grid, 256, 0, stream>>>(A, B, out);
}